// SparseSO3Conv_6708738916985
// MI455X (gfx1250) — compile-verified
//
#include <hip/hip_runtime.h>
#include <math.h>

typedef __attribute__((ext_vector_type(2))) float v2f;
typedef __attribute__((ext_vector_type(4))) float v4f;
typedef __attribute__((ext_vector_type(8))) float v8f;

#define WAVES 4                 // points per block, one wave per point
// LDS layout (float offsets)
#define O_H1   0                // 128 x 32
#define O_H2   4096             // 128 x 64
#define O_F    12288            // 128 x 8 (6 feats + zero pad)
#define O_CON  13312            // 4 x 64 contracted
#define O_W1F  13568            // 8x32 packed pairs
#define O_W2F  13824            // 32x64 packed pairs
#define O_W3F  15872            // 64x64 packed pairs
#define LDS_TOT 19968           // 79.9 KB

__device__ __forceinline__ v8f wmma4(v2f a, v2f b, v8f c) {
  return __builtin_amdgcn_wmma_f32_16x16x4_f32(false, a, false, b, (short)0, c,
                                               false, false);
}

__global__ __launch_bounds__(128) void so3conv_fused(
    const float* __restrict__ pts, const float* __restrict__ nbr,
    const float* __restrict__ w1,  const float* __restrict__ b1,
    const float* __restrict__ g1,  const float* __restrict__ be1,
    const float* __restrict__ w2,  const float* __restrict__ b2,
    const float* __restrict__ g2,  const float* __restrict__ be2,
    const float* __restrict__ w3,  const float* __restrict__ b3,
    const float* __restrict__ ow,  const float* __restrict__ ob,
    const float* __restrict__ lg,  const float* __restrict__ lb,
    float* __restrict__ out, int npts) {
  __shared__ __align__(16) float lds[LDS_TOT];
  const int tid  = threadIdx.x;
  const int lane = tid & 31;
  const int w    = tid >> 5;
  int gp = blockIdx.x * WAVES + w;
  if (gp >= npts) gp = npts - 1;      // uniform per wave: EXEC stays all-1s

  // ---- Stage weights into LDS, interleaved in B-fragment pair order ----
  // WF[((k>>1)*N + col)*2 + (k&1)] = W[k][col]
  for (int t = 0; t < 2; ++t) {        // W1 zero-padded to 8 rows
    int i = tid + t * 128, k = i >> 5, col = i & 31;
    float v = (k < 6) ? w1[k * 32 + col] : 0.f;
    lds[O_W1F + (((k >> 1) * 32 + col) << 1) + (k & 1)] = v;
  }
  for (int t = 0; t < 16; ++t) {       // W2 32x64
    int i = tid + t * 128, k = i >> 6, col = i & 63;
    lds[O_W2F + (((k >> 1) * 64 + col) << 1) + (k & 1)] = w2[i];
  }
  for (int t = 0; t < 32; ++t) {       // W3 64x64
    int i = tid + t * 128, k = i >> 6, col = i & 63;
    lds[O_W3F + (((k >> 1) * 64 + col) << 1) + (k & 1)] = w3[i];
  }

  // ---- Stage 0: rotation-invariant features ----
  {
    const float* nb = nbr + ((size_t)gp * 32 + (size_t)lane) * 3;
    float nx = nb[0], ny = nb[1], nz = nb[2];
    float mx = nx, my = ny, mz = nz;
    #pragma unroll
    for (int m = 1; m < 32; m <<= 1) {
      mx += __shfl_xor(mx, m, 32);
      my += __shfl_xor(my, m, 32);
      mz += __shfl_xor(mz, m, 32);
    }
    mx *= (1.f/32.f); my *= (1.f/32.f); mz *= (1.f/32.f);
    float px = pts[gp*3+0], py = pts[gp*3+1], pz = pts[gp*3+2];
    float l1x = mx-nx, l1y = my-ny, l1z = mz-nz;
    float l2x = nx-px, l2y = ny-py, l2z = nz-pz;
    float l3x = px-mx, l3y = py-my, l3z = pz-mz;
    float n1 = sqrtf(l1x*l1x + l1y*l1y + l1z*l1z);
    float n2 = sqrtf(l2x*l2x + l2y*l2y + l2z*l2z);
    float n3 = sqrtf(l3x*l3x + l3y*l3y + l3z*l3z);
    float t1 = (l1x*l2x + l1y*l2y + l1z*l2z) / (n1*n2 + 1e-7f);
    float t2 = (l2x*l3x + l2y*l3y + l2z*l3z) / (n2*n3 + 1e-7f);
    float t3 = (l3x*l1x + l3y*l1y + l3z*l1z) / (n3*n1 + 1e-7f);
    float* fr = &lds[O_F + tid * 8];
    fr[0]=n1; fr[1]=n2; fr[2]=n3; fr[3]=t1; fr[4]=t2; fr[5]=t3;
    fr[6]=0.f; fr[7]=0.f;
  }
  __syncthreads();

  const int M  = lane & 15;   // column within 16-lane half
  const int hi = lane >> 4;   // K offset select in A/B fragments

  // ---- Layer 1: 6(->8) -> 32 ----
  {
    v8f acc[2][2];
    #pragma unroll
    for (int rt = 0; rt < 2; ++rt)
      #pragma unroll
      for (int ct = 0; ct < 2; ++ct) {
        float bias = b1[ct*16 + M];
        #pragma unroll
        for (int i = 0; i < 8; ++i) acc[rt][ct][i] = bias;
      }
    #pragma unroll
    for (int s = 0; s < 2; ++s) {
      v2f a[2];
      #pragma unroll
      for (int rt = 0; rt < 2; ++rt)
        a[rt] = *(const v2f*)&lds[O_F + (w*32 + rt*16 + M)*8 + s*4 + 2*hi];
      #pragma unroll
      for (int ct = 0; ct < 2; ++ct) {
        v2f b = *(const v2f*)&lds[O_W1F + (((2*s + hi)*32 + ct*16 + M) << 1)];
        acc[0][ct] = wmma4(a[0], b, acc[0][ct]);
        acc[1][ct] = wmma4(a[1], b, acc[1][ct]);
      }
    }
    #pragma unroll
    for (int rt = 0; rt < 2; ++rt)
      #pragma unroll
      for (int r = 0; r < 8; ++r) {
        int rowr = w*32 + rt*16 + r + 8*hi;
        lds[O_H1 + rowr*32 +      M] = acc[rt][0][r];
        lds[O_H1 + rowr*32 + 16 + M] = acc[rt][1][r];
      }
  }
  __syncthreads();

  // ---- LN(32) + ReLU row pass: thread t owns row t ----
  {
    v4f x[8];
    float s = 0.f, q = 0.f;
    #pragma unroll
    for (int jj = 0; jj < 8; ++jj) {
      x[jj] = *(const v4f*)&lds[O_H1 + tid*32 + jj*4];
      #pragma unroll
      for (int c = 0; c < 4; ++c) { s += x[jj][c]; q += x[jj][c]*x[jj][c]; }
    }
    float mean = s * (1.f/32.f);
    float rstd = rsqrtf(q * (1.f/32.f) - mean*mean + 1e-5f);
    #pragma unroll
    for (int jj = 0; jj < 8; ++jj) {
      v4f y;
      #pragma unroll
      for (int c = 0; c < 4; ++c) {
        int j = jj*4 + c;                      // uniform -> scalar loads
        y[c] = fmaxf((x[jj][c]-mean)*rstd*g1[j] + be1[j], 0.f);
      }
      *(v4f*)&lds[O_H1 + tid*32 + jj*4] = y;
    }
  }
  __syncthreads();

  // ---- Layer 2: 32 -> 64 ----
  {
    v8f acc[2][4];
    #pragma unroll
    for (int rt = 0; rt < 2; ++rt)
      #pragma unroll
      for (int ct = 0; ct < 4; ++ct) {
        float bias = b2[ct*16 + M];
        #pragma unroll
        for (int i = 0; i < 8; ++i) acc[rt][ct][i] = bias;
      }
    #pragma unroll
    for (int s = 0; s < 8; ++s) {
      v2f a[2];
      #pragma unroll
      for (int rt = 0; rt < 2; ++rt)
        a[rt] = *(const v2f*)&lds[O_H1 + (w*32 + rt*16 + M)*32 + s*4 + 2*hi];
      #pragma unroll
      for (int ct = 0; ct < 4; ++ct) {
        v2f b = *(const v2f*)&lds[O_W2F + (((2*s + hi)*64 + ct*16 + M) << 1)];
        acc[0][ct] = wmma4(a[0], b, acc[0][ct]);
        acc[1][ct] = wmma4(a[1], b, acc[1][ct]);
      }
    }
    #pragma unroll
    for (int rt = 0; rt < 2; ++rt)
      #pragma unroll
      for (int r = 0; r < 8; ++r) {
        int rowr = w*32 + rt*16 + r + 8*hi;
        #pragma unroll
        for (int ct = 0; ct < 4; ++ct)
          lds[O_H2 + rowr*64 + ct*16 + M] = acc[rt][ct][r];
      }
  }
  __syncthreads();

  // ---- LN(64) + ReLU row pass ----
  {
    v4f x[16];
    float s = 0.f, q = 0.f;
    #pragma unroll
    for (int jj = 0; jj < 16; ++jj) {
      x[jj] = *(const v4f*)&lds[O_H2 + tid*64 + jj*4];
      #pragma unroll
      for (int c = 0; c < 4; ++c) { s += x[jj][c]; q += x[jj][c]*x[jj][c]; }
    }
    float mean = s * (1.f/64.f);
    float rstd = rsqrtf(q * (1.f/64.f) - mean*mean + 1e-5f);
    #pragma unroll
    for (int jj = 0; jj < 16; ++jj) {
      v4f y;
      #pragma unroll
      for (int c = 0; c < 4; ++c) {
        int j = jj*4 + c;
        y[c] = fmaxf((x[jj][c]-mean)*rstd*g2[j] + be2[j], 0.f);
      }
      *(v4f*)&lds[O_H2 + tid*64 + jj*4] = y;
    }
  }
  __syncthreads();

  // ---- Layer 3: 64 -> 64 (kern) + max over 32 neighbors ----
  {
    v8f acc[2][4];
    #pragma unroll
    for (int rt = 0; rt < 2; ++rt)
      #pragma unroll
      for (int ct = 0; ct < 4; ++ct) {
        float bias = b3[ct*16 + M];
        #pragma unroll
        for (int i = 0; i < 8; ++i) acc[rt][ct][i] = bias;
      }
    #pragma unroll
    for (int s = 0; s < 16; ++s) {
      v2f a[2];
      #pragma unroll
      for (int rt = 0; rt < 2; ++rt)
        a[rt] = *(const v2f*)&lds[O_H2 + (w*32 + rt*16 + M)*64 + s*4 + 2*hi];
      #pragma unroll
      for (int ct = 0; ct < 4; ++ct) {
        v2f b = *(const v2f*)&lds[O_W3F + (((2*s + hi)*64 + ct*16 + M) << 1)];
        acc[0][ct] = wmma4(a[0], b, acc[0][ct]);
        acc[1][ct] = wmma4(a[1], b, acc[1][ct]);
      }
    }
    #pragma unroll
    for (int ct = 0; ct < 4; ++ct) {
      float m = fmaxf(acc[0][ct][0], acc[1][ct][0]);
      #pragma unroll
      for (int r = 1; r < 8; ++r)
        m = fmaxf(m, fmaxf(acc[0][ct][r], acc[1][ct][r]));
      m = fmaxf(m, __shfl_xor(m, 16, 32));   // merge 16-lane halves
      if (hi == 0) lds[O_CON + w*64 + ct*16 + M] = m;
    }
  }
  __syncthreads();

  // ---- Output: contracted(64) @ ow(64x128) + b, LN(128) ----
  {
    v4f o = *(const v4f*)&ob[4*lane];        // lane owns features 4*lane..+3
    #pragma unroll
    for (int ii = 0; ii < 16; ++ii) {
      v4f c4 = *(const v4f*)&lds[O_CON + w*64 + ii*4];  // uniform LDS broadcast
      #pragma unroll
      for (int c = 0; c < 4; ++c) {
        const int i = ii*4 + c;
        v4f wv = *(const v4f*)&ow[i*128 + 4*lane];
        #pragma unroll
        for (int t = 0; t < 4; ++t) o[t] += c4[c] * wv[t];
      }
    }
    float s = 0.f, q = 0.f;
    #pragma unroll
    for (int t = 0; t < 4; ++t) { s += o[t]; q += o[t]*o[t]; }
    #pragma unroll
    for (int m = 1; m < 32; m <<= 1) {
      s += __shfl_xor(s, m, 32);
      q += __shfl_xor(q, m, 32);
    }
    float mean = s * (1.f/128.f);
    float rstd = rsqrtf(q * (1.f/128.f) - mean*mean + 1e-5f);
    v4f gv = *(const v4f*)&lg[4*lane];
    v4f bv = *(const v4f*)&lb[4*lane];
    v4f y;
    #pragma unroll
    for (int t = 0; t < 4; ++t) y[t] = (o[t]-mean)*rstd*gv[t] + bv[t];
    *(v4f*)&out[(size_t)gp*128 + 4*lane] = y;
  }
}

extern "C" void kernel_launch(void* const* d_in, const int* in_sizes, int n_in,
                              void* d_out, int out_size, void* d_ws, size_t ws_size,
                              hipStream_t stream) {
  (void)n_in; (void)out_size; (void)d_ws; (void)ws_size;
  const int npts = in_sizes[0] / 3;               // B*N
  const int blocks = (npts + WAVES - 1) / WAVES;  // one wave per point
  so3conv_fused<<<blocks, 32*WAVES, 0, stream>>>(
      (const float*)d_in[0],  (const float*)d_in[1],  (const float*)d_in[2],
      (const float*)d_in[3],  (const float*)d_in[4],  (const float*)d_in[5],
      (const float*)d_in[6],  (const float*)d_in[7],  (const float*)d_in[8],
      (const float*)d_in[9],  (const float*)d_in[10], (const float*)d_in[11],
      (const float*)d_in[12], (const float*)d_in[13], (const float*)d_in[14],
      (const float*)d_in[15], (float*)d_out, npts);
}